// GraphCNN_36636071035357
// MI455X (gfx1250) — compile-verified
//
#include <hip/hip_runtime.h>
#include <hip/hip_bf16.h>
#include <math.h>

#define NNODES 100000
#define INC    128
#define HIDC   64
#define OUTC   40
#define NEDGES 1000000
#define BN_EPS 1e-5f

typedef __attribute__((ext_vector_type(2))) float v2f;
typedef __attribute__((ext_vector_type(8))) float v8f;

// ---------- order-preserving float<->uint encoding for atomic max ----------
static __device__ __forceinline__ unsigned enc_f32(float f) {
    unsigned u = __float_as_uint(f);
    return (u & 0x80000000u) ? ~u : (u | 0x80000000u);
}
static __device__ __forceinline__ float dec_f32(unsigned e) {
    unsigned u = (e & 0x80000000u) ? (e ^ 0x80000000u) : ~e;
    return __uint_as_float(u);
}
#define ENC_NEG_INF 0x007FFFFFu   // enc_f32(-inf)

// ---------- fills (harness poisons d_ws/d_out; re-init every call) ----------
__global__ void fill_f32(float* __restrict__ p, float v, int n) {
    int i = blockIdx.x * blockDim.x + threadIdx.x;
    if (i < n) p[i] = v;
}
__global__ void fill_u32(unsigned* __restrict__ p, unsigned v, int n) {
    int i = blockIdx.x * blockDim.x + threadIdx.x;
    if (i < n) p[i] = v;
}

// ---------- fp32 WMMA GEMM: C[M,N] = A[M,K] * B[K,N], row major ----------
// One wave -> one 16x16 tile of C via fully unrolled V_WMMA_F32_16X16X4_F32 chain.
// Block stages the 16xK A tile and the whole KxN B (zero-padded to NPAD cols) in
// LDS once; MMA loop is branch-free (EXEC all-ones, as WMMA requires).
//   KP   = K+4 : A row stride -> 32 lanes' b64 reads cover 64 distinct banks
//   NPAD = 72  : 2*72 == 16 (mod 64) -> the two half-waves (rows k, k+2) use
//                disjoint bank ranges; also gives zero padding for N=40 tiles.
// ISA 7.12.2 operand layouts:
//   A 16x4:  lanes 0-15: M=lane,{K,K+1}; lanes 16-31: {K+2,K+3}
//   B 4x16:  vgpr r: lanes 0-15 row K=r, lanes 16-31 row K=r+2; col = lane&15
//   C/D:     vgpr v: lanes 0-15 M=v, lanes 16-31 M=v+8; col = lane&15
template <int K, int N, int NWAVES>
__global__ __launch_bounds__(32 * NWAVES)
void wmma_gemm_f32(const float* __restrict__ A,
                   const float* __restrict__ B,
                   float* __restrict__ C) {
    constexpr int KP   = K + 4;
    constexpr int NPAD = 72;
    __shared__ float sA[16 * KP];
    __shared__ float sB[K * NPAD];

    const int tid  = threadIdx.x;
    const int nthr = 32 * NWAVES;
    const int row0 = blockIdx.x * 16;

    // stage A tile (16 x K) with padded row stride, float4 chunks
    for (int i = tid; i < 16 * (K / 4); i += nthr) {
        int r = i / (K / 4), c4 = i % (K / 4);
        ((float4*)(sA + r * KP))[c4] = ((const float4*)(A + (size_t)(row0 + r) * K))[c4];
    }
    // stage B (K x N) into K x NPAD, zero-filling the pad columns
    for (int i = tid; i < K * NPAD; i += nthr) {
        int kk = i / NPAD, nn = i % NPAD;
        sB[i] = (nn < N) ? B[(size_t)kk * N + nn] : 0.f;
    }
    __syncthreads();

    const int lane  = tid & 31;
    const int wave  = tid >> 5;
    const int mrow  = lane & 15;
    const int khalf = (lane >> 4) << 1;      // 0 or 2
    const int n     = wave * 16 + mrow;      // output column (may exceed N-1: padded)

    v8f acc = {};
#pragma unroll
    for (int k = 0; k < K; k += 4) {
        float2 af = *(const float2*)(sA + mrow * KP + k + khalf);
        v2f a; a.x = af.x; a.y = af.y;
        v2f b;
        b.x = sB[(k + khalf)     * NPAD + n];
        b.y = sB[(k + khalf + 1) * NPAD + n];
        acc = __builtin_amdgcn_wmma_f32_16x16x4_f32(
            false, a, false, b, (short)0, acc, false, false);
    }
    const int rbase = row0 + ((lane >> 4) << 3);
    if (n < N) {
#pragma unroll
        for (int v = 0; v < 8; ++v)
            C[(size_t)(rbase + v) * N + n] = acc[v];
    }
}

// ---------- alpha_src / alpha_dst per node ----------
__global__ void attn_scores(const float* __restrict__ h,
                            const float* __restrict__ a_src,
                            const float* __restrict__ a_dst,
                            float* __restrict__ as, float* __restrict__ ad,
                            int N, int F) {
    int nidx = blockIdx.x * blockDim.x + threadIdx.x;
    if (nidx >= N) return;
    float s = 0.f, d = 0.f;
    const float* hp = h + (size_t)nidx * F;
    for (int f = 0; f < F; f += 4) {
        float4 v  = *(const float4*)(hp + f);
        float4 ws = *(const float4*)(a_src + f);
        float4 wd = *(const float4*)(a_dst + f);
        s += v.x * ws.x + v.y * ws.y + v.z * ws.z + v.w * ws.w;
        d += v.x * wd.x + v.y * wd.y + v.z * wd.z + v.w * wd.w;
    }
    as[nidx] = s; ad[nidx] = d;
}

static __device__ __forceinline__ void edge_sd(const long long* __restrict__ ei,
                                               int E, int e, int& s, int& d) {
    if (e < E) { s = (int)ei[e]; d = (int)ei[(size_t)E + e]; }
    else       { s = d = e - E; }            // self loops appended
}
static __device__ __forceinline__ float edge_logit(const float* __restrict__ as,
                                                   const float* __restrict__ ad,
                                                   int s, int d) {
    float x = as[s] + ad[d];
    return x > 0.f ? x : 0.2f * x;           // GAT leaky_relu slope
}

// ---------- edge pass 1: segment max ----------
__global__ void edge_max(const long long* __restrict__ ei,
                         const float* __restrict__ as, const float* __restrict__ ad,
                         unsigned* __restrict__ menc, int E, int ET) {
    int e = blockIdx.x * blockDim.x + threadIdx.x;
    if (e >= ET) return;
    int s, d; edge_sd(ei, E, e, s, d);
    atomicMax(&menc[d], enc_f32(edge_logit(as, ad, s, d)));
}

// ---------- edge pass 2: exp-sum ----------
__global__ void edge_expsum(const long long* __restrict__ ei,
                            const float* __restrict__ as, const float* __restrict__ ad,
                            const unsigned* __restrict__ menc,
                            float* __restrict__ denom, int E, int ET) {
    int e = blockIdx.x * blockDim.x + threadIdx.x;
    if (e >= ET) return;
    int s, d; edge_sd(ei, E, e, s, d);
    float ex = expf(edge_logit(as, ad, s, d) - dec_f32(menc[d]));
    atomicAdd(&denom[d], ex);
}

// ---------- edge pass 3: weighted scatter-add, one wave per edge ----------
// global_prefetch_b8 the gathered source row (and scatter target) right after
// index decode: hides the random L2 gather latency behind the scalar
// alpha computation (4 dependent loads + expf) that precedes the row loop.
__global__ void edge_aggregate(const long long* __restrict__ ei,
                               const float* __restrict__ as, const float* __restrict__ ad,
                               const unsigned* __restrict__ menc,
                               const float* __restrict__ denom,
                               const float* __restrict__ h,
                               float* __restrict__ out, int E, int ET, int F) {
    int lane = threadIdx.x & 31;
    int e = (int)(((size_t)blockIdx.x * blockDim.x + threadIdx.x) >> 5);
    if (e >= ET) return;
    int s, d; edge_sd(ei, E, e, s, d);
    const float* hs = h + (size_t)s * F;
    float* od       = out + (size_t)d * F;
    __builtin_prefetch(hs + lane, 0, 1);     // -> global_prefetch_b8 (read)
    __builtin_prefetch(od + lane, 1, 1);     // -> prefetch scatter target line
    float ex    = expf(edge_logit(as, ad, s, d) - dec_f32(menc[d]));
    float alpha = ex / (denom[d] + 1e-16f);
    for (int f = lane; f < F; f += 32)
        atomicAdd(&od[f], hs[f] * alpha);
}

// ---------- (+b1) -> BatchNorm(eval) -> leaky_relu(0.01) ----------
__global__ void bias_bn_leaky(const float* __restrict__ agg, const float* __restrict__ b,
                              const float* __restrict__ gamma, const float* __restrict__ beta,
                              const float* __restrict__ rmean, const float* __restrict__ rvar,
                              float* __restrict__ outp, int N, int F) {
    int i = blockIdx.x * blockDim.x + threadIdx.x;
    if (i >= N * F) return;
    int c = i % F;
    float v = agg[i] + b[c];
    v = (v - rmean[c]) * rsqrtf(rvar[c] + BN_EPS) * gamma[c] + beta[c];
    outp[i] = v > 0.f ? v : 0.01f * v;
}

// ---------- log_softmax + masked NLL accumulation ----------
__global__ void logsoftmax_loss(const float* __restrict__ agg,
                                const float* __restrict__ b,
                                const long long* __restrict__ y,
                                const unsigned char* __restrict__ mask,
                                float* __restrict__ out,
                                float* __restrict__ lossacc, int N) {
    int n = blockIdx.x * blockDim.x + threadIdx.x;
    if (n >= N) return;
    float buf[OUTC];
    const float* ap = agg + (size_t)n * OUTC;
    float mx = -INFINITY;
#pragma unroll
    for (int c = 0; c < OUTC; ++c) { buf[c] = ap[c] + b[c]; mx = fmaxf(mx, buf[c]); }
    float sum = 0.f;
#pragma unroll
    for (int c = 0; c < OUTC; ++c) sum += expf(buf[c] - mx);
    float lse = mx + logf(sum);
    float* op = out + (size_t)n * OUTC;
#pragma unroll
    for (int c = 0; c < OUTC; ++c) op[c] = buf[c] - lse;
    float mk = mask[n] ? 1.f : 0.f;
    int cls = (int)y[n];
    atomicAdd(&lossacc[0], -(buf[cls] - lse) * mk);
    atomicAdd(&lossacc[1], mk);
}

__global__ void finalize_loss(const float* __restrict__ acc, float* __restrict__ loss_out) {
    loss_out[0] = acc[0] / fmaxf(acc[1], 1.f);
}

extern "C" void kernel_launch(void* const* d_in, const int* in_sizes, int n_in,
                              void* d_out, int out_size, void* d_ws, size_t ws_size,
                              hipStream_t stream) {
    const float*         x        = (const float*)d_in[0];
    const long long*     ei       = (const long long*)d_in[1];
    const long long*     y        = (const long long*)d_in[2];
    const unsigned char* mask     = (const unsigned char*)d_in[3];
    const float*         W1       = (const float*)d_in[4];
    const float*         att_s1   = (const float*)d_in[5];
    const float*         att_d1   = (const float*)d_in[6];
    const float*         b1       = (const float*)d_in[7];
    const float*         gamma    = (const float*)d_in[8];
    const float*         beta     = (const float*)d_in[9];
    const float*         rmean    = (const float*)d_in[10];
    const float*         rvar     = (const float*)d_in[11];
    const float*         W2       = (const float*)d_in[12];
    const float*         att_s2   = (const float*)d_in[13];
    const float*         att_d2   = (const float*)d_in[14];
    const float*         b2       = (const float*)d_in[15];

    float* out = (float*)d_out;                       // [N*OUTC] logsoftmax, then loss scalar
    const int N = NNODES, ET = NEDGES + NNODES;

    // ---- workspace carve ----
    float* ws = (float*)d_ws;
    size_t o = 0;
    float*    h1     = ws + o; o += (size_t)N * HIDC;   // hidden features; reused as hid post-BN
    float*    agg1   = ws + o; o += (size_t)N * HIDC;
    float*    h2     = ws + o; o += (size_t)N * OUTC;
    float*    agg2   = ws + o; o += (size_t)N * OUTC;
    float*    as1    = ws + o; o += N;
    float*    ad1    = ws + o; o += N;
    float*    as2    = ws + o; o += N;
    float*    ad2    = ws + o; o += N;
    float*    den1   = ws + o; o += N;
    float*    den2   = ws + o; o += N;
    unsigned* m1     = (unsigned*)(ws + o); o += N;
    unsigned* m2     = (unsigned*)(ws + o); o += N;
    float*    lacc   = ws + o; o += 2;

    const int T = 256;
    // ---- init scratch (poisoned by harness) ----
    fill_f32<<<(N * HIDC + T - 1) / T, T, 0, stream>>>(agg1, 0.f, N * HIDC);
    fill_f32<<<(N * OUTC + T - 1) / T, T, 0, stream>>>(agg2, 0.f, N * OUTC);
    fill_f32<<<(N + T - 1) / T, T, 0, stream>>>(den1, 0.f, N);
    fill_f32<<<(N + T - 1) / T, T, 0, stream>>>(den2, 0.f, N);
    fill_u32<<<(N + T - 1) / T, T, 0, stream>>>(m1, ENC_NEG_INF, N);
    fill_u32<<<(N + T - 1) / T, T, 0, stream>>>(m2, ENC_NEG_INF, N);
    fill_f32<<<1, 2, 0, stream>>>(lacc, 0.f, 2);

    const int eb  = (ET + T - 1) / T;          // thread-per-edge blocks
    const int ewb = (ET + 7) / 8;              // wave-per-edge blocks (8 waves / 256 thr)

    // ---- layer 1: GAT(128 -> 64) ----
    wmma_gemm_f32<INC, HIDC, 4><<<N / 16, 128, 0, stream>>>(x, W1, h1);
    attn_scores<<<(N + T - 1) / T, T, 0, stream>>>(h1, att_s1, att_d1, as1, ad1, N, HIDC);
    edge_max   <<<eb,  T, 0, stream>>>(ei, as1, ad1, m1, NEDGES, ET);
    edge_expsum<<<eb,  T, 0, stream>>>(ei, as1, ad1, m1, den1, NEDGES, ET);
    edge_aggregate<<<ewb, T, 0, stream>>>(ei, as1, ad1, m1, den1, h1, agg1, NEDGES, ET, HIDC);
    bias_bn_leaky<<<(N * HIDC + T - 1) / T, T, 0, stream>>>(agg1, b1, gamma, beta, rmean, rvar,
                                                            h1 /* hid (reuse) */, N, HIDC);

    // ---- layer 2: GAT(64 -> 40) ----
    wmma_gemm_f32<HIDC, OUTC, 3><<<N / 16, 96, 0, stream>>>(h1, W2, h2);
    attn_scores<<<(N + T - 1) / T, T, 0, stream>>>(h2, att_s2, att_d2, as2, ad2, N, OUTC);
    edge_max   <<<eb,  T, 0, stream>>>(ei, as2, ad2, m2, NEDGES, ET);
    edge_expsum<<<eb,  T, 0, stream>>>(ei, as2, ad2, m2, den2, NEDGES, ET);
    edge_aggregate<<<ewb, T, 0, stream>>>(ei, as2, ad2, m2, den2, h2, agg2, NEDGES, ET, OUTC);

    // ---- log_softmax + loss ----
    logsoftmax_loss<<<(N + T - 1) / T, T, 0, stream>>>(agg2, b2, y, mask, out, lacc, N);
    finalize_loss<<<1, 1, 0, stream>>>(lacc, out + (size_t)N * OUTC);
}